// BaseSearchBasedModel_39608188404354
// MI455X (gfx1250) — compile-verified
//
#include <hip/hip_runtime.h>

typedef __attribute__((ext_vector_type(2))) float v2f;
typedef __attribute__((ext_vector_type(4))) float v4f;
typedef __attribute__((ext_vector_type(8))) float v8f;

#define BATCH  2048
#define LSEQ   1024
#define KTOP   128
#define DD     64
#define HH     4
#define STRIDE 65   // padded LDS row stride (floats) -> conflict-free A-frag reads

// One 64x64 projection of topk[128x64] with bias, via V_WMMA_F32_16X16X4_F32.
// wave w: ntile = w&3 ; mtiles = (w<4 ? 0..3 : 4..7). All 8 waves participate.
__device__ __forceinline__ void proj_wmma(const float* __restrict__ W,
                                          const float* __restrict__ bias,
                                          const float* __restrict__ topk,
                                          float* __restrict__ dst,
                                          int wave, int lane)
{
    const int ntile   = wave & 3;
    const int mbase   = (wave < 4) ? 0 : 4;
    const int lcol    = lane & 15;
    const int halfsel = (lane < 16) ? 0 : 2;      // K-pair selector (ISA 7.12.2)
    const int col     = ntile * 16 + lcol;

    // B-matrix fragments: B is 4x16 (KxN); lane<16 -> K={0,1}, lane>=16 -> K={2,3}
    v2f bf[16];
#pragma unroll
    for (int kk = 0; kk < 16; ++kk) {
        bf[kk].x = W[(kk * 4 + halfsel)     * DD + col];
        bf[kk].y = W[(kk * 4 + halfsel + 1) * DD + col];
    }
    const float bv = bias[col];

    for (int mt = mbase; mt < mbase + 4; ++mt) {
        v8f acc = {};
        // A-matrix 16x4 f32: lane<16 row M=lane cols {k,k+1}; lane>=16 cols {k+2,k+3}
        const float* arow = topk + (mt * 16 + lcol) * STRIDE + halfsel;
#pragma unroll
        for (int kk = 0; kk < 16; ++kk) {
            v2f a;
            a.x = arow[kk * 4];
            a.y = arow[kk * 4 + 1];
            acc = __builtin_amdgcn_wmma_f32_16x16x4_f32(
                false, a, false, bf[kk], (short)0, acc, false, false);
        }
        // D layout: VGPR i -> row mt*16 + i (+8 for upper half-lanes), N = col
        const int rbase = mt * 16 + ((lane < 16) ? 0 : 8);
#pragma unroll
        for (int i = 0; i < 8; ++i)
            dst[(rbase + i) * STRIDE + col] = acc[i] + bv;
    }
}

__global__ __launch_bounds__(256) void din_fused_kernel(
    const int*   __restrict__ indices,   // [B,K]
    const float* __restrict__ seq,       // [B,L,D]
    const float* __restrict__ tgt_g,     // [B,1,D]
    const float* __restrict__ WQ, const float* __restrict__ bQ,
    const float* __restrict__ WK, const float* __restrict__ bK,
    const float* __restrict__ WV, const float* __restrict__ bV,
    const float* __restrict__ WO, const float* __restrict__ bO,
    const float* __restrict__ W1, const float* __restrict__ b1,
    const float* __restrict__ W2, const float* __restrict__ b2,
    float* __restrict__ out)             // [B,1]
{
    extern __shared__ float lds[];
    float* topk  = lds;                       // 128*65
    float* kv    = topk + KTOP * STRIDE;      // 128*65 (K-mat then V-mat, reused)
    float* tgt   = kv + KTOP * STRIDE;        // 64
    float* qv    = tgt + DD;                  // 4*64 scaled queries (bias incl.)
    float* sc    = qv + HH * DD;              // 128 raw scores
    float* pr    = sc + KTOP;                 // 128 unnormalized exp()
    float* heads = pr + KTOP;                 // 4*64
    float* mh    = heads + HH * DD;           // 64
    float* red   = mh + DD;                   // 64

    const int tid  = threadIdx.x;
    const int lane = tid & 31;
    const int wave = tid >> 5;
    const int b    = blockIdx.x;

    // ---- target row ----
    if (tid < DD) tgt[tid] = tgt_g[b * DD + tid];

    // ---- gather top-K history rows into padded LDS (float4 global reads) ----
    for (int i = tid; i < KTOP * 16; i += 256) {
        const int row = i >> 4;
        const int c4  = (i & 15) << 2;
        const int idx = indices[b * KTOP + row];
        const float* src = seq + ((size_t)b * LSEQ + (size_t)idx) * DD + c4;
        v4f val = *(const v4f*)src;
        float* dstp = topk + row * STRIDE + c4;
        dstp[0] = val.x; dstp[1] = val.y; dstp[2] = val.z; dstp[3] = val.w;
    }
    __syncthreads();

    // ---- all 4 head queries at once; fold 1/sqrt(D)=0.125 into q ----
    {
        const int h = tid >> 6;
        const int e = tid & 63;
        const float* wq = WQ + h * DD * DD;
        float acc = bQ[h * DD + e];
#pragma unroll 8
        for (int d0 = 0; d0 < DD; ++d0)
            acc += tgt[d0] * wq[d0 * DD + e];
        qv[h * DD + e] = acc * 0.125f;
    }
    __syncthreads();

    for (int h = 0; h < HH; ++h) {
        // ===== K projection (WMMA) -> kv =====
        proj_wmma(WK + h * DD * DD, bK + h * DD, topk, kv, wave, lane);
        __syncthreads();

        // ===== scores = q . k_row =====
        if (tid < KTOP) {
            const float* q    = qv + h * DD;
            const float* krow = kv + tid * STRIDE;
            float s = 0.f;
#pragma unroll 8
            for (int d0 = 0; d0 < DD; ++d0) s += q[d0] * krow[d0];
            sc[tid] = s;
        }
        __syncthreads();

        // ===== exp(s - max) (threads 0..127) overlapped with V projection =====
        if (tid < KTOP) {
            float m = -3.0e38f;
            for (int j = 0; j < KTOP; ++j) m = fmaxf(m, sc[j]);
            pr[tid] = __expf(sc[tid] - m);
        }
        // kv already consumed by the score pass -> safe to overwrite
        proj_wmma(WV + h * DD * DD, bV + h * DD, topk, kv, wave, lane);
        __syncthreads();

        // ===== heads[h] = softmax(attn) . V  (normalization folded in) =====
        if (tid < DD) {
            float S = 0.f, acc = 0.f;
            for (int k = 0; k < KTOP; ++k) {
                const float p = pr[k];
                S   += p;
                acc += p * kv[k * STRIDE + tid];
            }
            heads[h * DD + tid] = acc / S;
        }
        __syncthreads();
    }

    // ---- output projection: mhta = flat(heads) @ WO + bO ----
    if (tid < DD) {
        float acc = bO[tid];
#pragma unroll 4
        for (int i = 0; i < HH * DD; ++i)
            acc += heads[i] * WO[i * DD + tid];
        mh[tid] = acc;
    }
    __syncthreads();

    // ---- MLP: relu([mhta, tgt] @ W1 + b1) then dot with W2 ----
    if (tid < DD) {
        float acc = b1[tid];
#pragma unroll 8
        for (int i = 0; i < DD; ++i) acc += mh[i]  * W1[i * DD + tid];
#pragma unroll 8
        for (int i = 0; i < DD; ++i) acc += tgt[i] * W1[(DD + i) * DD + tid];
        red[tid] = fmaxf(acc, 0.f) * W2[tid];
    }
    __syncthreads();

    if (tid == 0) {
        float s = b2[0];
        for (int j = 0; j < DD; ++j) s += red[j];
        out[b] = s;
    }
}

extern "C" void kernel_launch(void* const* d_in, const int* in_sizes, int n_in,
                              void* d_out, int out_size, void* d_ws, size_t ws_size,
                              hipStream_t stream) {
    const int*   indices = (const int*)  d_in[0];
    const float* seq     = (const float*)d_in[1];
    const float* tgt     = (const float*)d_in[2];
    const float* WQ = (const float*)d_in[3];
    const float* bQ = (const float*)d_in[4];
    const float* WK = (const float*)d_in[5];
    const float* bK = (const float*)d_in[6];
    const float* WV = (const float*)d_in[7];
    const float* bV = (const float*)d_in[8];
    const float* WO = (const float*)d_in[9];
    const float* bO = (const float*)d_in[10];
    const float* W1 = (const float*)d_in[11];
    const float* b1 = (const float*)d_in[12];
    const float* W2 = (const float*)d_in[13];
    const float* b2 = (const float*)d_in[14];
    float* out = (float*)d_out;

    const size_t ldsFloats = 2 * KTOP * STRIDE + DD + HH * DD + KTOP + KTOP +
                             HH * DD + DD + DD;
    const size_t ldsBytes = ldsFloats * sizeof(float);   // ~68.8 KB

    hipFuncSetAttribute(reinterpret_cast<const void*>(din_fused_kernel),
                        hipFuncAttributeMaxDynamicSharedMemorySize,
                        (int)ldsBytes);

    din_fused_kernel<<<BATCH, 256, ldsBytes, stream>>>(
        indices, seq, tgt, WQ, bQ, WK, bK, WV, bV, WO, bO, W1, b1, W2, b2, out);
}